// GATESG_70600672411978
// MI455X (gfx1250) — compile-verified
//
#include <hip/hip_runtime.h>
#include <math.h>

// GAT 2-layer forward for MI455X (gfx1250, wave32).
// GEMMs use native fp32 WMMA (v_wmma_f32_16x16x4_f32) -> full reference precision.
// Edge softmax/aggregation uses f32 atomics with float4 gathers; all hot buffers
// (h1: 51MB, agg1: 51MB, stats) fit in the 192MB L2 -> atomics/gathers at L2 rate.

typedef __attribute__((ext_vector_type(2))) float v2f;
typedef __attribute__((ext_vector_type(8))) float v8f;

#define TPB 256

// ---------------------------------------------------------------- utilities
__global__ void fill_kernel(float* __restrict__ p, float v, int n) {
  int i = blockIdx.x * blockDim.x + threadIdx.x;
  if (i < n) p[i] = v;
}

// sign-aware float atomic max via int/uint ordering trick (init value must be -inf)
__device__ __forceinline__ void atomicMaxF(float* addr, float v) {
  if (v >= 0.0f) {
    atomicMax((int*)addr, __float_as_int(v));
  } else {
    atomicMin((unsigned int*)addr, __float_as_uint(v));
  }
}

// ---------------------------------------------------------------- WMMA GEMM
// C[rows x cols] = A[rows x K] @ B[K x cols].
// grid = (rows/16, cols/64), block = 32 (one wave). Each wave: 16x64 strip,
// 4 accumulator tiles, A fragment reused across the 4 WMMA issues per k-step.
// fp32 WMMA fragment layout (ISA 7.12.2): A: lane holds row M=lane&15,
// VGPR0/1 = K = 2*(lane>=16) and K+1. B mirrored with N on lanes.
// C/D: VGPR r -> row r (+8 for lanes>=16), col = lane&15.
__global__ __launch_bounds__(32)
void gemm_wmma_f32(const float* __restrict__ A, const float* __restrict__ B,
                   float* __restrict__ C, int K, int cols) {
  const int rowBase = blockIdx.x << 4;
  const int colBase = blockIdx.y << 6;   // 64 cols per wave
  const int lane = threadIdx.x;
  const int m  = lane & 15;
  const int kh = (lane >> 4) << 1;       // 0 or 2
  const float* __restrict__ arow = A + (size_t)(rowBase + m) * K + kh;
  const float* __restrict__ bp   = B + colBase + m;
  v8f acc0 = {}, acc1 = {}, acc2 = {}, acc3 = {};
  #pragma unroll 2
  for (int k0 = 0; k0 < K; k0 += 4) {
    v2f a;
    a.x = arow[k0];
    a.y = arow[k0 + 1];
    const float* __restrict__ br0 = bp + (size_t)(k0 + kh) * cols;
    const float* __restrict__ br1 = bp + (size_t)(k0 + kh + 1) * cols;
    v2f b;
    b.x = br0[0];  b.y = br1[0];
    acc0 = __builtin_amdgcn_wmma_f32_16x16x4_f32(false, a, false, b, (short)0, acc0, false, false);
    b.x = br0[16]; b.y = br1[16];
    acc1 = __builtin_amdgcn_wmma_f32_16x16x4_f32(false, a, false, b, (short)0, acc1, false, false);
    b.x = br0[32]; b.y = br1[32];
    acc2 = __builtin_amdgcn_wmma_f32_16x16x4_f32(false, a, false, b, (short)0, acc2, false, false);
    b.x = br0[48]; b.y = br1[48];
    acc3 = __builtin_amdgcn_wmma_f32_16x16x4_f32(false, a, false, b, (short)0, acc3, false, false);
  }
  const int rowOff = rowBase + ((lane >> 4) << 3);
  #pragma unroll
  for (int r = 0; r < 8; ++r) {
    size_t base = (size_t)(rowOff + r) * cols + colBase + m;
    C[base]      = acc0[r];
    C[base + 16] = acc1[r];
    C[base + 32] = acc2[r];
    C[base + 48] = acc3[r];
  }
}

// ---------------------------------------------------------------- per-node alphas
// h layout [N, H, F]; id = n*H + h. alpha_src[id] = h[n,h,:] . a_src[h,:]
__global__ void node_alpha_kernel(const float* __restrict__ h,
                                  const float* __restrict__ avs,
                                  const float* __restrict__ avd,
                                  float* __restrict__ as, float* __restrict__ ad,
                                  int NH, int H, int F) {
  int id = blockIdx.x * blockDim.x + threadIdx.x;
  if (id >= NH) return;
  int hh = id % H;                       // H is 1 or 2
  const float4* __restrict__ hp = (const float4*)(h + (size_t)id * F);
  const float4* __restrict__ sv = (const float4*)(avs + hh * F);
  const float4* __restrict__ dv = (const float4*)(avd + hh * F);
  float accS = 0.f, accD = 0.f;
  for (int f = 0; f < F / 4; ++f) {
    float4 hv = hp[f], s4 = sv[f], d4 = dv[f];
    accS += hv.x * s4.x + hv.y * s4.y + hv.z * s4.z + hv.w * s4.w;
    accD += hv.x * d4.x + hv.y * d4.y + hv.z * d4.z + hv.w * d4.w;
  }
  as[id] = accS;
  ad[id] = accD;
}

// ---------------------------------------------------------------- edge passes
// edges 0..E-1 from edge_index (int64, [2,E] flat: src=ei[e], dst=ei[E+e]);
// edges E..EE-1 are self-loops (src=dst=e-E). One thread per edge, loop heads.
__global__ void edge_logits_max(const long long* __restrict__ ei, int E, int EE,
                                const float* __restrict__ as, const float* __restrict__ ad,
                                float* __restrict__ ebuf, float* __restrict__ mbuf, int H) {
  int e = blockIdx.x * blockDim.x + threadIdx.x;
  if (e >= EE) return;
  int src, dst;
  if (e < E) { src = (int)ei[e]; dst = (int)ei[(size_t)E + e]; }
  else       { src = e - E; dst = src; }
  for (int hh = 0; hh < H; ++hh) {
    float v = as[src * H + hh] + ad[dst * H + hh];
    v = (v > 0.f) ? v : 0.2f * v;                 // leaky_relu(0.2)
    ebuf[(size_t)e * H + hh] = v;
    atomicMaxF(&mbuf[dst * H + hh], v);
  }
}

__global__ void edge_exp_sum(const long long* __restrict__ ei, int E, int EE,
                             const float* __restrict__ mbuf,
                             float* __restrict__ ebuf, float* __restrict__ sbuf, int H) {
  int e = blockIdx.x * blockDim.x + threadIdx.x;
  if (e >= EE) return;
  int dst = (e < E) ? (int)ei[(size_t)E + e] : (e - E);
  for (int hh = 0; hh < H; ++hh) {
    float p = __expf(ebuf[(size_t)e * H + hh] - mbuf[dst * H + hh]);
    ebuf[(size_t)e * H + hh] = p;
    atomicAdd(&sbuf[dst * H + hh], p);
  }
}

// HF = H*F features per node. HF/4 threads per edge, float4 gathers
// (global_load_b128), 4 scalar f32 atomic adds per thread (no packed f32 atomic).
// Multiple edges per 256-thread block: 4 (HF=256) or 16 (HF=64).
template <int HF, int H>
__global__ __launch_bounds__(TPB)
void edge_aggregate(const long long* __restrict__ ei, int E, int EE,
                    const float* __restrict__ h,
                    const float* __restrict__ p,
                    const float* __restrict__ s,
                    float* __restrict__ agg) {
  constexpr int TPE = HF / 4;                  // threads per edge
  const int e = blockIdx.x * (TPB / TPE) + threadIdx.x / TPE;
  if (e >= EE) return;
  const int t = (threadIdx.x % TPE) * 4;       // feature offset (aligned 16B)
  const int hh = (H == 1) ? 0 : t / (HF / H);  // head of this 4-feature chunk
  int src, dst;
  if (e < E) { src = (int)ei[e]; dst = (int)ei[(size_t)E + e]; }
  else       { src = e - E; dst = src; }
  const float alpha = p[(size_t)e * H + hh] / s[(size_t)dst * H + hh];
  const float4 hv = *(const float4*)(h + (size_t)src * HF + t);
  float* __restrict__ ap = agg + (size_t)dst * HF + t;
  atomicAdd(ap + 0, alpha * hv.x);
  atomicAdd(ap + 1, alpha * hv.y);
  atomicAdd(ap + 2, alpha * hv.z);
  atomicAdd(ap + 3, alpha * hv.w);
}

// ---------------------------------------------------------------- epilogues
__global__ void elu_bias_inplace(float* __restrict__ a, const float* __restrict__ b,
                                 int n, int cmask) {
  int i = blockIdx.x * blockDim.x + threadIdx.x;
  if (i >= n) return;
  float v = a[i] + b[i & cmask];
  a[i] = (v > 0.f) ? v : expm1f(v);
}

__global__ void add_bias(const float* __restrict__ a, const float* __restrict__ b,
                         float* __restrict__ out, int n, int cmask) {
  int i = blockIdx.x * blockDim.x + threadIdx.x;
  if (i >= n) return;
  out[i] = a[i] + b[i & cmask];
}

// ---------------------------------------------------------------- launch
extern "C" void kernel_launch(void* const* d_in, const int* in_sizes, int n_in,
                              void* d_out, int out_size, void* d_ws, size_t ws_size,
                              hipStream_t stream) {
  const float*     x      = (const float*)d_in[0];
  const long long* ei     = (const long long*)d_in[1];   // int64 [2,E]
  const float*     W1     = (const float*)d_in[2];
  const float*     a_src1 = (const float*)d_in[3];
  const float*     a_dst1 = (const float*)d_in[4];
  const float*     b1     = (const float*)d_in[5];
  const float*     W2     = (const float*)d_in[6];
  const float*     a_src2 = (const float*)d_in[7];
  const float*     a_dst2 = (const float*)d_in[8];
  const float*     b2     = (const float*)d_in[9];
  float* out = (float*)d_out;

  const int IN_DIM = 256, C1 = 256, C2 = 64, H1 = 2, F1 = 128;
  const int N  = in_sizes[0] / IN_DIM;   // 50000 (multiple of 16)
  const int E  = in_sizes[1] / 2;        // 800000
  const int EE = E + N;                  // + self loops

  // workspace layout (floats)
  float* ws = (float*)d_ws;
  size_t off = 0;
  auto alloc = [&](size_t n) { float* p = ws + off; off += n; return p; };
  float* h1raw = alloc((size_t)N * C1);   // x @ W1
  float* agg1  = alloc((size_t)N * C1);   // layer1 aggregation -> elu in place
  float* h2raw = alloc((size_t)N * C2);   // h1 @ W2
  float* agg2  = alloc((size_t)N * C2);
  float* as1 = alloc((size_t)N * H1);
  float* ad1 = alloc((size_t)N * H1);
  float* m1  = alloc((size_t)N * H1);
  float* s1  = alloc((size_t)N * H1);
  float* as2 = alloc(N);
  float* ad2 = alloc(N);
  float* m2  = alloc(N);
  float* s2  = alloc(N);
  float* eb1 = alloc((size_t)EE * H1);
  float* eb2 = alloc((size_t)EE);
  (void)ws_size; (void)n_in; (void)out_size;

  auto blocks = [](int n) { return (n + TPB - 1) / TPB; };
  const float NEG_INF = -__builtin_huge_valf();

  // init accumulators / stats
  fill_kernel<<<blocks(N * H1), TPB, 0, stream>>>(m1, NEG_INF, N * H1);
  fill_kernel<<<blocks(N * H1), TPB, 0, stream>>>(s1, 0.f, N * H1);
  fill_kernel<<<blocks(N * C1), TPB, 0, stream>>>(agg1, 0.f, N * C1);
  fill_kernel<<<blocks(N), TPB, 0, stream>>>(m2, NEG_INF, N);
  fill_kernel<<<blocks(N), TPB, 0, stream>>>(s2, 0.f, N);
  fill_kernel<<<blocks(N * C2), TPB, 0, stream>>>(agg2, 0.f, N * C2);

  // ---- layer 1 ----
  gemm_wmma_f32<<<dim3(N / 16, C1 / 64), 32, 0, stream>>>(x, W1, h1raw, IN_DIM, C1);
  node_alpha_kernel<<<blocks(N * H1), TPB, 0, stream>>>(h1raw, a_src1, a_dst1, as1, ad1,
                                                        N * H1, H1, F1);
  edge_logits_max<<<blocks(EE), TPB, 0, stream>>>(ei, E, EE, as1, ad1, eb1, m1, H1);
  edge_exp_sum<<<blocks(EE), TPB, 0, stream>>>(ei, E, EE, m1, eb1, s1, H1);
  {
    constexpr int EPB = TPB / (256 / 4);   // 4 edges per block
    edge_aggregate<256, 2><<<(EE + EPB - 1) / EPB, TPB, 0, stream>>>(
        ei, E, EE, h1raw, eb1, s1, agg1);
  }
  elu_bias_inplace<<<blocks(N * C1), TPB, 0, stream>>>(agg1, b1, N * C1, C1 - 1);

  // ---- layer 2 ----
  gemm_wmma_f32<<<dim3(N / 16, C2 / 64), 32, 0, stream>>>(agg1, W2, h2raw, C1, C2);
  node_alpha_kernel<<<blocks(N), TPB, 0, stream>>>(h2raw, a_src2, a_dst2, as2, ad2,
                                                   N, 1, C2);
  edge_logits_max<<<blocks(EE), TPB, 0, stream>>>(ei, E, EE, as2, ad2, eb2, m2, 1);
  edge_exp_sum<<<blocks(EE), TPB, 0, stream>>>(ei, E, EE, m2, eb2, s2, 1);
  {
    constexpr int EPB = TPB / (64 / 4);    // 16 edges per block
    edge_aggregate<64, 1><<<(EE + EPB - 1) / EPB, TPB, 0, stream>>>(
        ei, E, EE, h2raw, eb2, s2, agg2);
  }
  add_bias<<<blocks(N * C2), TPB, 0, stream>>>(agg2, b2, out, N * C2, C2 - 1);
}